// PSMLayer_83777632076060
// MI455X (gfx1250) — compile-verified
//
#include <hip/hip_runtime.h>

// PSMLayer: out = (W0 @ W1 @ W2 @ W3 @ U.T).T + bias
// Four NT-GEMM stages (Out[b,d] = sum_k W[d,k]*X[b,k]), each storing its
// result transposed into a [B,D] row-major buffer so stages chain directly
// and the last one emits y.T + bias.
//
// MI455X: 256 MB fp32 weights stream from HBM each call -> memory-bound
// (~14 us floor @ 23.3 TB/s). fp32 V_WMMA_F32_16X16X4_F32 gives ample
// compute at full precision; the optimization target is the memory pipe:
// async global->LDS copies (ASYNCcnt) + LDS double buffering.

typedef float v2f __attribute__((ext_vector_type(2)));
typedef float v4f __attribute__((ext_vector_type(4)));
typedef float v8f __attribute__((ext_vector_type(8)));
typedef int   v4i __attribute__((ext_vector_type(4)));

#define DD 4096      // factor dim (rows of W, k-dim)
#define BB 2048      // batch rows of U / X
#define BM 128       // block tile along D
#define BN 128       // block tile along B
#define KB 32        // k-tile depth per LDS stage
#define LSTR 36      // LDS row stride (floats): 16B-aligned rows (144 B),
                     // conflict-free across 16 rows (36*dr % 64 != 0, dr<16)
#define NT (DD / KB)

#if defined(__HIP_DEVICE_COMPILE__) &&                                   \
    __has_builtin(__builtin_amdgcn_global_load_async_to_lds_b128) &&     \
    __has_builtin(__builtin_amdgcn_s_wait_asynccnt)
#define USE_ASYNC 1
#else
#define USE_ASYNC 0
#endif

#if USE_ASYNC
typedef __attribute__((address_space(1))) v4i gv4i;  // global int4
typedef __attribute__((address_space(3))) v4i lv4i;  // LDS int4

__device__ __forceinline__ void cp_async16(const float* g, float* l) {
  __builtin_amdgcn_global_load_async_to_lds_b128(
      (gv4i*)(g), (lv4i*)(l), /*offset=*/0, /*cpol=*/0);
}
#endif

__global__ __launch_bounds__(256) void psm_gemm_nt(
    const float* __restrict__ W,    // [DD, DD] row-major
    const float* __restrict__ X,    // [BB, DD] row-major (k contiguous)
    float* __restrict__ Out,        // [BB, DD]: Out[b,d] = sum_k W[d,k]*X[b,k]
    const float* __restrict__ bias) // [DD] or nullptr
{
#if USE_ASYNC
  __shared__ float sW[2 * BM * LSTR];   // double-buffered: 2 x 18 KB
  __shared__ float sX[2 * BN * LSTR];
#else
  __shared__ float sW[BM * LSTR];
  __shared__ float sX[BN * LSTR];
#endif

  const int tid   = threadIdx.x;
  const int lane  = tid & 31;     // wave32
  const int wave  = tid >> 5;     // 8 waves
  const int waveM = wave & 1;     // 2 waves along M -> 128
  const int waveN = wave >> 1;    // 4 waves along N -> 128
  const int m16   = lane & 15;
  const int khalf = lane >> 4;    // lanes 16-31 hold K+2 (16x4 f32 A layout)

  const int blockM = blockIdx.y * BM;  // D strip (W rows / output d)
  const int blockN = blockIdx.x * BN;  // B strip (X rows / output b)

  // fill pattern: 256 threads x 4 passes, one float4 each -> 128 rows x 32 cols
  const int fillR0 = tid >> 3;         // 0..31
  const int fillC  = (tid & 7) * 4;    // 0..28

  v8f acc[4][2];
#pragma unroll
  for (int i = 0; i < 4; ++i)
#pragma unroll
    for (int j = 0; j < 2; ++j) acc[i][j] = (v8f)0.0f;

  auto compute = [&](const float* bW, const float* bX) {
#pragma unroll
    for (int kk = 0; kk < KB; kk += 4) {
      v2f a[4], b[2];
#pragma unroll
      for (int i = 0; i < 4; ++i)
        a[i] = *(const v2f*)&bW[(waveM * 64 + i * 16 + m16) * LSTR + kk + 2 * khalf];
#pragma unroll
      for (int j = 0; j < 2; ++j)
        b[j] = *(const v2f*)&bX[(waveN * 32 + j * 16 + m16) * LSTR + kk + 2 * khalf];
#pragma unroll
      for (int i = 0; i < 4; ++i)
#pragma unroll
        for (int j = 0; j < 2; ++j)
          acc[i][j] = __builtin_amdgcn_wmma_f32_16x16x4_f32(
              false, a[i], false, b[j], (short)0, acc[i][j], false, false);
    }
  };

#if USE_ASYNC
  auto issue = [&](int t, int buf) {
    const int kb = t * KB;
    float* dW = &sW[buf * (BM * LSTR)];
    float* dX = &sX[buf * (BN * LSTR)];
#pragma unroll
    for (int p = 0; p < 4; ++p) {
      const int r = fillR0 + p * 32;
      cp_async16(W + (size_t)(blockM + r) * DD + kb + fillC, &dW[r * LSTR + fillC]);
      cp_async16(X + (size_t)(blockN + r) * DD + kb + fillC, &dX[r * LSTR + fillC]);
    }
  };

  issue(0, 0);
  for (int t = 0; t < NT; ++t) {
    const int buf = t & 1;
    __builtin_amdgcn_s_wait_asynccnt(0);  // own copies for tile t complete
    __syncthreads();                      // everyone's copies visible; prev
                                          // reads of buf^1 finished
    if (t + 1 < NT) issue(t + 1, buf ^ 1);
    compute(&sW[buf * (BM * LSTR)], &sX[buf * (BN * LSTR)]);
  }
#else
  // register-prefetch double buffering: keep all 8 tile loads in flight.
  v4f wreg[4], xreg[4];
  auto loadg = [&](int t, v4f* wr, v4f* xr) {
    const int kb = t * KB;
#pragma unroll
    for (int p = 0; p < 4; ++p) {
      const int r = fillR0 + p * 32;
      wr[p] = *(const v4f*)(W + (size_t)(blockM + r) * DD + kb + fillC);
      xr[p] = *(const v4f*)(X + (size_t)(blockN + r) * DD + kb + fillC);
    }
  };
  loadg(0, wreg, xreg);
  for (int t = 0; t < NT; ++t) {
    __syncthreads();  // prev compute's LDS reads done
#pragma unroll
    for (int p = 0; p < 4; ++p) {
      const int r = fillR0 + p * 32;
      *(v4f*)&sW[r * LSTR + fillC] = wreg[p];
      *(v4f*)&sX[r * LSTR + fillC] = xreg[p];
    }
    v4f wn[4], xn[4];
    loadg(t + 1 < NT ? t + 1 : t, wn, xn);  // issue before compute (overlap)
    __syncthreads();
    compute(sW, sX);
#pragma unroll
    for (int p = 0; p < 4; ++p) { wreg[p] = wn[p]; xreg[p] = xn[p]; }
  }
#endif

  // Epilogue: C layout (16x16 f32): VGPR r -> M = r + 8*khalf, N = m16.
  // Transposed store as two b128 per tile (+ v4f bias).
#pragma unroll
  for (int i = 0; i < 4; ++i) {
#pragma unroll
    for (int j = 0; j < 2; ++j) {
      const int nG = blockN + waveN * 32 + j * 16 + m16;
      const int mG = blockM + waveM * 64 + i * 16 + khalf * 8;
      v8f a8 = acc[i][j];
      v4f lo = __builtin_shufflevector(a8, a8, 0, 1, 2, 3);
      v4f hi = __builtin_shufflevector(a8, a8, 4, 5, 6, 7);
      if (bias) {
        lo += *(const v4f*)(bias + mG);
        hi += *(const v4f*)(bias + mG + 4);
      }
      *(v4f*)(Out + (size_t)nG * DD + mG)     = lo;
      *(v4f*)(Out + (size_t)nG * DD + mG + 4) = hi;
    }
  }
}

extern "C" void kernel_launch(void* const* d_in, const int* in_sizes, int n_in,
                              void* d_out, int out_size, void* d_ws, size_t ws_size,
                              hipStream_t stream) {
  const float* U    = (const float*)d_in[0];  // [BB, DD]
  const float* W0   = (const float*)d_in[1];  // [DD, DD]
  const float* W1   = (const float*)d_in[2];
  const float* W2   = (const float*)d_in[3];
  const float* W3   = (const float*)d_in[4];
  const float* bias = (const float*)d_in[5];  // [DD]

  float* out = (float*)d_out;
  float* ws0 = (float*)d_ws;  // one [BB, DD] scratch (32 MB)

  dim3 grid(BB / BN, DD / BM);  // x = N tiles (reuse W strip in L2), y = M strips
  dim3 block(256);

  psm_gemm_nt<<<grid, block, 0, stream>>>(W3, U,   ws0, nullptr);
  psm_gemm_nt<<<grid, block, 0, stream>>>(W2, ws0, out, nullptr);
  psm_gemm_nt<<<grid, block, 0, stream>>>(W1, out, ws0, nullptr);
  psm_gemm_nt<<<grid, block, 0, stream>>>(W0, ws0, out, bias);
}